// _QConv2d_SLOW_26817775796835
// MI455X (gfx1250) — compile-verified
//
#include <hip/hip_runtime.h>
#include <math.h>

typedef __attribute__((ext_vector_type(2))) float v2f;
typedef __attribute__((ext_vector_type(8))) float v8f;

#define HDIM 128
#define WDIM 128
#define NBATCH 16
#define NCH 16
#define TILES_W (WDIM / 16)                       // 8
#define TILES_TOTAL (NBATCH * HDIM * TILES_W)     // 16384

// ---------------------------------------------------------------------------
// Kernel 1: build the 16x16 circuit unitary once (1 wave, 16 active threads).
// Writes Ur (256 floats) then Ui (256 floats), row-major [row][col], to ws.
// ---------------------------------------------------------------------------
__global__ __launch_bounds__(32) void build_unitary_kernel(
    const float* __restrict__ wts,   // [2, 4, 3]
    float* __restrict__ ug)          // [512] workspace
{
    __shared__ float Ur[16][16];
    __shared__ float Ui[16][16];

    const int t = threadIdx.x;
    if (t < 16) {
        for (int r = 0; r < 16; ++r) {
            Ur[r][t] = (r == t) ? 1.0f : 0.0f;
            Ui[r][t] = 0.0f;
        }
        for (int l = 0; l < 2; ++l) {
            // Rot(phi, theta, omega) on each wire (wire i <-> bit (3-i), MSB first)
            for (int i = 0; i < 4; ++i) {
                float phi = wts[(l * 4 + i) * 3 + 0];
                float th  = wts[(l * 4 + i) * 3 + 1];
                float om  = wts[(l * 4 + i) * 3 + 2];
                float c  = cosf(0.5f * th), s = sinf(0.5f * th);
                float A  = -0.5f * (phi + om), B = -0.5f * (phi - om);
                float cA = cosf(A), sA = sinf(A);
                float cB = cosf(B), sB = sinf(B);
                // U00 = ep*c, U01 = -conj(em)*s, U10 = em*s, U11 = conj(ep)*c
                float u00r =  c * cA, u00i =  c * sA;
                float u01r = -s * cB, u01i =  s * sB;
                float u10r =  s * cB, u10i =  s * sB;
                float u11r =  c * cA, u11i = -c * sA;
                int mask = 1 << (3 - i);
                for (int j = 0; j < 16; ++j) {
                    if (j & mask) continue;
                    int j1 = j | mask;
                    float x0r = Ur[j][t],  x0i = Ui[j][t];
                    float x1r = Ur[j1][t], x1i = Ui[j1][t];
                    Ur[j][t]  = u00r * x0r - u00i * x0i + u01r * x1r - u01i * x1i;
                    Ui[j][t]  = u00r * x0i + u00i * x0r + u01r * x1i + u01i * x1r;
                    Ur[j1][t] = u10r * x0r - u10i * x0i + u11r * x1r - u11i * x1i;
                    Ui[j1][t] = u10r * x0i + u10i * x0r + u11r * x1i + u11i * x1r;
                }
            }
            // Ring of CNOTs, range r = l % 3 + 1
            int rr = (l % 3) + 1;
            for (int i = 0; i < 4; ++i) {
                int cmask = 1 << (3 - i);
                int tmask = 1 << (3 - ((i + rr) & 3));
                for (int j = 0; j < 16; ++j) {
                    if ((j & cmask) && !(j & tmask)) {
                        int j1 = j | tmask;
                        float tr = Ur[j][t]; Ur[j][t] = Ur[j1][t]; Ur[j1][t] = tr;
                        float ti = Ui[j][t]; Ui[j][t] = Ui[j1][t]; Ui[j1][t] = ti;
                    }
                }
            }
        }
        // Thread t only reads the column it wrote: no barrier needed.
        for (int r = 0; r < 16; ++r) {
            ug[r * 16 + t]       = Ur[r][t];
            ug[256 + r * 16 + t] = Ui[r][t];
        }
    }
}

// ---------------------------------------------------------------------------
// Kernel 2: bandwidth-bound main pass. One wave per 16-pixel tile iteration:
//   A (16x4 f32) = rows of Ur/Ui, B (4x16) = per-pixel amplitudes,
//   D[chan][pix] -> coalesced 64B half-wave stores per accumulator component.
// ---------------------------------------------------------------------------
__global__ __launch_bounds__(256) void qconv_wmma_kernel(
    const float* __restrict__ x,     // [16, 1, 128, 128]
    const float* __restrict__ ug,    // [512] unitary (Ur | Ui)
    float* __restrict__ out)         // [16, 16, 128, 128]
{
    const int tid  = threadIdx.x;
    const int lane = tid & 31;
    const int wid  = tid >> 5;          // wave id in block: 0..7
    const int col  = lane & 15;         // M/N index this lane serves
    const bool hi  = lane >= 16;        // upper half-wave: K+2,K+3 slice
    const int koff = hi ? 2 : 0;

    // Preload A operand (rows of U) for the 4 K-chunks, real & imag.
    // 16 L2-resident loads per lane, once per kernel.
    v2f Ar[4], Ai[4];
#pragma unroll
    for (int kc = 0; kc < 4; ++kc) {
        int base = col * 16 + 4 * kc + koff;
        Ar[kc].x = ug[base + 0];
        Ar[kc].y = ug[base + 1];
        Ai[kc].x = ug[256 + base + 0];
        Ai[kc].y = ug[256 + base + 1];
    }

    const int waves_total = gridDim.x * 8;
    for (int tile = blockIdx.x * 8 + wid; tile < TILES_TOTAL; tile += waves_total) {
        int b  = tile / (HDIM * TILES_W);
        int rm = tile - b * (HDIM * TILES_W);
        int h  = rm / TILES_W;
        int w0 = (rm - h * TILES_W) * 16;
        int wp = w0 + col;              // this lane's pixel (both half-waves)

        const float* xb = x + (size_t)b * (HDIM * WDIM);

        // 3x3 patch taps (+0.01, zero-padded borders -> 0.01).
        // Branchless: clamped in-bounds load + cndmask.
        float tap[9];
#pragma unroll
        for (int dy = -1; dy <= 1; ++dy) {
#pragma unroll
            for (int dx = -1; dx <= 1; ++dx) {
                int hy = h + dy, wx = wp + dx;
                bool ok = (hy >= 0) & (hy < HDIM) & (wx >= 0) & (wx < WDIM);
                int hyc = min(max(hy, 0), HDIM - 1);
                int wxc = min(max(wx, 0), WDIM - 1);
                float v = xb[hyc * WDIM + wxc];
                v = ok ? v : 0.0f;
                tap[(dy + 1) * 3 + (dx + 1)] = v + 0.01f;
            }
        }
        float n2 = 7.0f * 1.0e-4f;      // seven 0.01 pad entries
#pragma unroll
        for (int k = 0; k < 9; ++k) n2 += tap[k] * tap[k];
        float inv = 1.0f / sqrtf(n2);

        float amp[16];
#pragma unroll
        for (int k = 0; k < 9; ++k) amp[k] = tap[k] * inv;
        float qpad = 0.01f * inv;
#pragma unroll
        for (int k = 9; k < 16; ++k) amp[k] = qpad;

        v8f cr = {0.f, 0.f, 0.f, 0.f, 0.f, 0.f, 0.f, 0.f};
        v8f ci = {0.f, 0.f, 0.f, 0.f, 0.f, 0.f, 0.f, 0.f};
#pragma unroll
        for (int kc = 0; kc < 4; ++kc) {
            v2f Bk;   // B column-major per lane; constant indices -> cndmask, no scratch
            Bk.x = hi ? amp[4 * kc + 2] : amp[4 * kc + 0];
            Bk.y = hi ? amp[4 * kc + 3] : amp[4 * kc + 1];
            cr = __builtin_amdgcn_wmma_f32_16x16x4_f32(false, Ar[kc], false, Bk,
                                                       (short)0, cr, false, false);
            ci = __builtin_amdgcn_wmma_f32_16x16x4_f32(false, Ai[kc], false, Bk,
                                                       (short)0, ci, false, false);
        }

        // D[chan][pix]: lane col = pixel, component v = channel (v / v+8 for hi half).
        // Per component: lanes 0-15 write 16 contiguous floats -> coalesced.
        int cbase = hi ? 8 : 0;
        float* ob = out + (size_t)b * NCH * HDIM * WDIM + (size_t)h * WDIM + w0 + col;
#pragma unroll
        for (int v = 0; v < 8; ++v) {
            float pr = cr[v], pi = ci[v];
            float prob = pr * pr + pi * pi;
            float o = fminf(prob * 8.0f, 1.0f);   // prob >= 0, lower clip free
            ob[(size_t)(cbase + v) * (HDIM * WDIM)] = o;
        }
    }
}

extern "C" void kernel_launch(void* const* d_in, const int* in_sizes, int n_in,
                              void* d_out, int out_size, void* d_ws, size_t ws_size,
                              hipStream_t stream) {
    const float* x   = (const float*)d_in[0];   // [16,1,128,128] f32
    const float* wts = (const float*)d_in[1];   // [2,4,3] f32
    float* out = (float*)d_out;                 // [16,16,128,128] f32
    float* ug  = (float*)d_ws;                  // 512 floats (2 KB) of scratch

    build_unitary_kernel<<<1, 32, 0, stream>>>(wts, ug);
    // 1024 blocks x 8 waves = 8192 waves; 16384 tiles -> 2 tiles/wave.
    qconv_wmma_kernel<<<1024, 256, 0, stream>>>(x, ug, out);
}